// DenseMaskPredictor_8358006358544
// MI455X (gfx1250) — compile-verified
//
#include <hip/hip_runtime.h>
#include <hip/hip_bf16.h>

typedef __attribute__((ext_vector_type(2))) float v2f;
typedef __attribute__((ext_vector_type(8))) float v8f;

#define NN 128
#define CC 80
#define MM 28
#define HH 768
#define WW 768
#define NUMVALID 80
#define ROWS_PER_WG 64
#define STRIPS (HH / ROWS_PER_WG)   // 12
#define TSTRIDE 772                  // 768 + 4 pad: de-conflict LDS banks between K-rows

__global__ __launch_bounds__(256) void paste_wmma_kernel(
    const float* __restrict__ mask,     // [N, C, 28, 28]
    const int*   __restrict__ clsidx,   // [N]
    const float* __restrict__ bbox,     // [N, 4] (x0, y0, x1, y1)
    float*       __restrict__ out)      // [N, 768, 768]
{
    __shared__ float Tlds[32 * TSTRIDE];           // T = P @ Wx^T, padded to 32 rows
    __shared__ float p[MM][32];                    // sigmoid(mask) (or 0 if invalid)
    __shared__ float wx0s[WW], wx1s[WW];
    __shared__ int   ix0s[WW], ix1s[WW];
    __shared__ float wy0s[ROWS_PER_WG], wy1s[ROWS_PER_WG];
    __shared__ int   iy0s[ROWS_PER_WG], iy1s[ROWS_PER_WG];

    const int wg     = blockIdx.x;
    const int n      = wg / STRIPS;
    const int strip  = wg % STRIPS;
    const int ybase0 = strip * ROWS_PER_WG;
    const int tid    = threadIdx.x;

    const float bx0 = bbox[n * 4 + 0];
    const float by0 = bbox[n * 4 + 1];
    const float bx1 = bbox[n * 4 + 2];
    const float by1 = bbox[n * 4 + 3];
    const int   ci    = clsidx[n];
    const bool  valid = (ci >= 0) && (ci < NUMVALID);
    const int   cc    = min(max(ci, 0), CC - 1);

    // ---- per-x interpolation tables (768 entries) ----
    const float sx = (float)MM / (bx1 - bx0);
    for (int x = tid; x < WW; x += 256) {
        float gx = ((float)x + 0.5f - bx0) * sx - 0.5f;
        float f  = floorf(gx);
        int   i0 = (int)f;
        int   i1 = i0 + 1;
        float w1 = gx - f;
        float w0 = 1.0f - w1;
        w0 = (i0 >= 0 && i0 < MM) ? w0 : 0.0f;
        w1 = (i1 >= 0 && i1 < MM) ? w1 : 0.0f;
        ix0s[x] = min(max(i0, 0), MM - 1);
        ix1s[x] = min(max(i1, 0), MM - 1);
        wx0s[x] = w0;
        wx1s[x] = w1;
    }

    // ---- per-y interpolation tables (64 entries for this strip) ----
    const float sy = (float)MM / (by1 - by0);
    if (tid < ROWS_PER_WG) {
        float gy = ((float)(ybase0 + tid) + 0.5f - by0) * sy - 0.5f;
        float f  = floorf(gy);
        int   i0 = (int)f;
        int   i1 = i0 + 1;
        float w1 = gy - f;
        float w0 = 1.0f - w1;
        w0 = (i0 >= 0 && i0 < MM) ? w0 : 0.0f;
        w1 = (i1 >= 0 && i1 < MM) ? w1 : 0.0f;
        iy0s[tid] = min(max(i0, 0), MM - 1);
        iy1s[tid] = min(max(i1, 0), MM - 1);
        wy0s[tid] = w0;
        wy1s[tid] = w1;
    }

    // ---- P = sigmoid(mask[n, cc]) (zero if invalid class) ----
    const float* msrc = mask + ((size_t)n * CC + cc) * (MM * MM);
    for (int e = tid; e < MM * MM; e += 256) {
        int r = e / MM, c = e % MM;
        float v = msrc[e];
        p[r][c] = valid ? (1.0f / (1.0f + __expf(-v))) : 0.0f;
    }
    __syncthreads();

    // ---- stage 1: T[i][x] = wx0*P[i][ix0] + wx1*P[i][ix1]  (rows 28..31 zero) ----
    for (int e = tid; e < 32 * WW; e += 256) {
        int i = e / WW, x = e % WW;
        float t = 0.0f;
        if (i < MM)
            t = wx0s[x] * p[i][ix0s[x]] + wx1s[x] * p[i][ix1s[x]];
        Tlds[i * TSTRIDE + x] = t;
    }
    __syncthreads();

    // ---- stage 2: out_tile = Wy_tile(16x32) @ T(32x16) via V_WMMA_F32_16X16X4_F32 ----
    const int wave    = tid >> 5;        // 0..7
    const int lane    = tid & 31;
    const int rowtile = wave >> 1;       // 0..3 (16 rows each)
    const int laneM   = lane & 15;
    const int hi      = lane >> 4;       // half-wave: K offset +2

    // Build A (16x32, 8 chunks of 16x4). A row = Wy row: <=2 nonzeros.
    const int   yl  = rowtile * 16 + laneM;
    const int   iy0 = iy0s[yl], iy1 = iy1s[yl];
    const float wy0 = wy0s[yl], wy1 = wy1s[yl];
    v2f a[8];
#pragma unroll
    for (int k = 0; k < 8; ++k) {
        int kb = 4 * k + 2 * hi;
        float a0 = ((kb     == iy0) ? wy0 : 0.0f) + ((kb     == iy1) ? wy1 : 0.0f);
        float a1 = ((kb + 1 == iy0) ? wy0 : 0.0f) + ((kb + 1 == iy1) ? wy1 : 0.0f);
        a[k].x = a0;
        a[k].y = a1;
    }

    const size_t outbase = (size_t)n * ((size_t)HH * WW);
    for (int ct = (wave & 1); ct < WW / 16; ct += 2) {
        const int xb = ct * 16;
        v8f acc = {};
#pragma unroll
        for (int k = 0; k < 8; ++k) {
            int kb = 4 * k + 2 * hi;
            v2f b;
            b.x = Tlds[kb * TSTRIDE + xb + laneM];
            b.y = Tlds[(kb + 1) * TSTRIDE + xb + laneM];
            acc = __builtin_amdgcn_wmma_f32_16x16x4_f32(
                false, a[k], false, b, (short)0, acc, false, false);
        }
        const int yg = ybase0 + rowtile * 16 + 8 * hi;   // C/D: VGPR r -> row r (+8 for hi half)
        float* op = out + outbase + (size_t)yg * WW + xb + laneM;
#pragma unroll
        for (int r = 0; r < 8; ++r) {
            __builtin_nontemporal_store(acc[r], op + (size_t)r * WW);
        }
    }
}

extern "C" void kernel_launch(void* const* d_in, const int* in_sizes, int n_in,
                              void* d_out, int out_size, void* d_ws, size_t ws_size,
                              hipStream_t stream) {
    const float* mask   = (const float*)d_in[0];  // [128,80,28,28]
    const int*   cls    = (const int*)d_in[1];    // [128]
    const float* bbox   = (const float*)d_in[2];  // [128,4]
    (void)d_in; (void)in_sizes; (void)n_in; (void)d_ws; (void)ws_size; (void)out_size;

    float* out = (float*)d_out;                   // [128,768,768]
    dim3 grid(NN * STRIPS);                       // 128 * 12 = 1536 workgroups
    dim3 block(256);                              // 8 wave32
    paste_wmma_kernel<<<grid, block, 0, stream>>>(mask, cls, bbox, out);
}